// Decoder_3272765079948
// MI455X (gfx1250) — compile-verified
//
#include <hip/hip_runtime.h>
#include <math.h>

#define B_   256
#define P_   196
#define ENC_ 2048
#define EMB_ 512
#define DEC_ 512
#define ATT_ 512
#define V_   10000
#define L_   21
#define T_   20
#define XD_  (EMB_ + ENC_)   // 2560

typedef float v2f __attribute__((ext_vector_type(2)));
typedef float v8f __attribute__((ext_vector_type(8)));

__device__ __forceinline__ float sigf(float x){ return 1.0f/(1.0f + __expf(-x)); }

// ---------------- stable descending sort by caption length ----------------
__global__ void sort_kernel(const int* __restrict__ cap_len,
                            int* __restrict__ sort_ind, int* __restrict__ dec_len){
  __shared__ int ls[B_];
  int i = threadIdx.x;
  ls[i] = cap_len[i];                 // caption_lengths[:,0]
  __syncthreads();
  int li = ls[i];
  int rank = 0;
  for (int j = 0; j < B_; ++j){
    int lj = ls[j];
    rank += (lj > li) || (lj == li && j < i);
  }
  sort_ind[rank] = i;
  dec_len[rank]  = li - 1;
}

__global__ void rowmap_kernel(const int* __restrict__ sort_ind, int* __restrict__ rowmap){
  int idx = blockIdx.x*256 + threadIdx.x;
  if (idx >= B_*P_) return;
  int b = idx / P_, p = idx - b*P_;
  rowmap[idx] = sort_ind[b]*P_ + p;
}

// ---------------- mean over P of gathered encoder rows ----------------
__global__ void mean_kernel(const float* __restrict__ enc, const int* __restrict__ sort_ind,
                            float* __restrict__ mean_enc){
  int b = blockIdx.y;
  int e = blockIdx.x*256 + threadIdx.x;
  long base = (long)sort_ind[b]*P_*ENC_ + e;
  float acc = 0.f;
  for (int p = 0; p < P_; ++p) acc += enc[base + (long)p*ENC_];
  mean_enc[(long)b*ENC_ + e] = acc * (1.0f/(float)P_);
}

// ---------------- generic f32 WMMA GEMM: Out = A(MxK) * W(NxK)^T + bias ----------------
// Tiles staged with GLOBAL_LOAD_ASYNC_TO_LDS_B128 (ASYNCcnt), double-buffered so the
// next tile's HBM fetch overlaps the current tile's v_wmma_f32_16x16x4_f32 work.
// MODE 0: C init = bias, plain store
// MODE 1: C init = Out + bias (accumulate pass)
// MODE 2: C init = bias, masked strided store (vocab projection -> d_out)
#define BM 64
#define BN 64
#define BK 16
#define LPAD 20   // row stride in floats: 80B (16B-aligned rows, conflict-free frag reads)

template<int MODE>
__global__ __launch_bounds__(256) void gemm_wmma_f32(
    const float* __restrict__ A, const float* __restrict__ W,
    const float* __restrict__ bias, float* __restrict__ Out,
    int M, int N, int K, long ors, long oco,
    const int* __restrict__ rowmap, const int* __restrict__ dec_len, int t)
{
  __shared__ __align__(16) float As[2][BM][LPAD];  // [buf][m][k]
  __shared__ __align__(16) float Bs[2][BN][LPAD];  // [buf][n][k]  (W layout; frag reads transpose)

  const int tid  = threadIdx.x;
  const int lane = tid & 31;
  const int wave = tid >> 5;
  const int m0 = blockIdx.y * BM;
  const int n0 = blockIdx.x * BN;

  const int wm  = wave & 3;          // wave's 16-row tile within block
  const int wn2 = wave >> 2;         // wave's 32-col half within block
  const int tm  = wm * 16;
  const int tn  = wn2 * 32;
  const int lm  = lane & 15;
  const int hi  = lane >> 4;

  const int nA = n0 + tn + lm;       // lane's N column (tile 0)
  const int nB = nA + 16;            // lane's N column (tile 1)

  v8f c0, c1;
  const float bv0 = (nA < N) ? bias[nA] : 0.f;
  const float bv1 = (nB < N) ? bias[nB] : 0.f;
  #pragma unroll
  for (int r = 0; r < 8; ++r){
    if (MODE == 1){
      long m = m0 + tm + r + 8*hi;
      c0[r] = Out[m*ors + oco + nA] + bv0;
      c1[r] = Out[m*ors + oco + nB] + bv1;
    } else {
      c0[r] = bv0; c1[r] = bv1;
    }
  }

  // per-thread staging slots: one 16B async copy each into A-tile and B-tile
  const int arow = tid >> 2;                   // 0..63
  const int akq  = (tid & 3) * 4;              // k offset within chunk (floats)
  const long asrc = rowmap ? (long)rowmap[m0 + arow] : (long)(m0 + arow);
  const float* Ag = A + asrc*(long)K + akq;
  const int bsrc = (n0 + arow < N) ? (n0 + arow) : (N - 1);  // clamp: cols >= N never stored
  const float* Wg = W + (long)bsrc*K + akq;

  auto stage = [&](int k0, int buf){
    unsigned la = (unsigned)(uintptr_t)&As[buf][arow][akq];
    unsigned lb = (unsigned)(uintptr_t)&Bs[buf][arow][akq];
    unsigned long long ga = (unsigned long long)(uintptr_t)(Ag + k0);
    unsigned long long gb = (unsigned long long)(uintptr_t)(Wg + k0);
    asm volatile("global_load_async_to_lds_b128 %0, %1, off" :: "v"(la), "v"(ga) : "memory");
    asm volatile("global_load_async_to_lds_b128 %0, %1, off" :: "v"(lb), "v"(gb) : "memory");
  };

  stage(0, 0);
  int cur = 0;
  for (int k0 = 0; k0 < K; k0 += BK){
    const bool more = (k0 + BK) < K;
    if (more){
      stage(k0 + BK, cur ^ 1);
      asm volatile("s_wait_asynccnt 2" ::: "memory");  // this chunk done, next may fly
    } else {
      asm volatile("s_wait_asynccnt 0" ::: "memory");
    }
    __syncthreads();
    const float (*Ab)[LPAD] = As[cur];
    const float (*Bb)[LPAD] = Bs[cur];
    #pragma unroll
    for (int kk = 0; kk < BK; kk += 4){
      // ISA 7.12.2 f32 frags: A lane=(lane&15, 2*hi+v); B lane=(2*hi+v, lane&15)
      v2f a, b0, b1;
      a[0]  = Ab[tm + lm][kk + 2*hi + 0];
      a[1]  = Ab[tm + lm][kk + 2*hi + 1];
      b0[0] = Bb[tn + lm][kk + 2*hi + 0];
      b0[1] = Bb[tn + lm][kk + 2*hi + 1];
      b1[0] = Bb[tn + 16 + lm][kk + 2*hi + 0];
      b1[1] = Bb[tn + 16 + lm][kk + 2*hi + 1];
      c0 = __builtin_amdgcn_wmma_f32_16x16x4_f32(false, a, false, b0, (short)0, c0, false, false);
      c1 = __builtin_amdgcn_wmma_f32_16x16x4_f32(false, a, false, b1, (short)0, c1, false, false);
    }
    __syncthreads();          // buffer 'cur' free for reuse by next stage()
    cur ^= 1;
  }

  #pragma unroll
  for (int r = 0; r < 8; ++r){
    long m = m0 + tm + r + 8*hi;
    float v0 = c0[r], v1 = c1[r];
    if (MODE == 2){
      bool act = (t < dec_len[m]);
      v0 = act ? v0 : 0.f;
      v1 = act ? v1 : 0.f;
    }
    if (nA < N) Out[m*ors + oco + nA] = v0;
    if (nB < N) Out[m*ors + oco + nB] = v1;
  }
}

// ---------------- scores: relu(att1 + att2) . w_full + b_full ----------------
__global__ void scores_kernel(const float* __restrict__ att1, const float* __restrict__ att2,
                              const float* __restrict__ wfa, const float* __restrict__ bfa,
                              float* __restrict__ scores){
  int gw   = blockIdx.x * 8 + (threadIdx.x >> 5);   // (b,p) index, 0..B*P-1
  int lane = threadIdx.x & 31;
  int b = gw / P_;
  const float* a1 = att1 + (long)gw * ATT_;
  const float* a2 = att2 + (long)b  * ATT_;
  float acc = 0.f;
  #pragma unroll 4
  for (int j = lane; j < ATT_; j += 32){
    float v = a1[j] + a2[j];
    acc += fmaxf(v, 0.f) * wfa[j];
  }
  #pragma unroll
  for (int off = 16; off; off >>= 1) acc += __shfl_xor(acc, off, 32);
  if (lane == 0) scores[gw] = acc + bfa[0];
}

// ---------------- softmax over P per batch row ----------------
__global__ void softmax_kernel(const float* __restrict__ scores, float* __restrict__ alpha){
  __shared__ float red[256];
  int b = blockIdx.x, i = threadIdx.x;
  float s = (i < P_) ? scores[b*P_ + i] : -3.4e38f;
  red[i] = s; __syncthreads();
  for (int off = 128; off; off >>= 1){
    if (i < off) red[i] = fmaxf(red[i], red[i+off]);
    __syncthreads();
  }
  float mx = red[0]; __syncthreads();
  float e = (i < P_) ? __expf(s - mx) : 0.f;
  red[i] = e; __syncthreads();
  for (int off = 128; off; off >>= 1){
    if (i < off) red[i] += red[i+off];
    __syncthreads();
  }
  float inv = 1.0f / red[0];
  if (i < P_) alpha[b*P_ + i] = e * inv;
}

// ---------------- awe[b,e] = sum_p alpha[b,p] * enc[sort(b),p,e] ----------------
__global__ void awe_kernel(const float* __restrict__ enc, const float* __restrict__ alpha,
                           const int* __restrict__ sort_ind, float* __restrict__ awe){
  __shared__ float al[P_];
  int b = blockIdx.y;
  if (threadIdx.x < P_) al[threadIdx.x] = alpha[b*P_ + threadIdx.x];
  __syncthreads();
  int e = blockIdx.x*256 + threadIdx.x;
  long base = (long)sort_ind[b]*P_*ENC_ + e;
  float acc = 0.f;
  for (int p = 0; p < P_; ++p) acc += al[p] * enc[base + (long)p*ENC_];
  awe[(long)b*ENC_ + e] = acc;
}

// ---------------- x = [ emb[cap_t] | sigmoid(gpre)*awe ] ----------------
__global__ void build_x_kernel(const int* __restrict__ caps, const int* __restrict__ sort_ind,
                               const float* __restrict__ emb, const float* __restrict__ gpre,
                               const float* __restrict__ awe, float* __restrict__ x, int t){
  int b = blockIdx.y;
  int j = blockIdx.x*256 + threadIdx.x;
  long xb = (long)b*XD_;
  if (j < EMB_){
    int cid = caps[sort_ind[b]*L_ + t];
    x[xb + j] = emb[(long)cid*EMB_ + j];
  } else {
    int e = j - EMB_;
    x[xb + j] = sigf(gpre[(long)b*ENC_ + e]) * awe[(long)b*ENC_ + e];
  }
}

// ---------------- LSTM pointwise with active mask ----------------
__global__ void lstm_kernel(const float* __restrict__ gates, float* __restrict__ h,
                            float* __restrict__ c, const int* __restrict__ dec_len, int t){
  int b = blockIdx.y;
  int d = blockIdx.x*256 + threadIdx.x;
  long gb = (long)b*4*DEC_;
  float ig = sigf (gates[gb + d]);
  float fg = sigf (gates[gb + DEC_   + d]);
  float gg = tanhf(gates[gb + 2*DEC_ + d]);
  float og = sigf (gates[gb + 3*DEC_ + d]);
  if (t < dec_len[b]){
    long idx = (long)b*DEC_ + d;
    float cn = fg * c[idx] + ig * gg;
    c[idx] = cn;
    h[idx] = og * tanhf(cn);
  }
}

extern "C" void kernel_launch(void* const* d_in, const int* in_sizes, int n_in,
                              void* d_out, int out_size, void* d_ws, size_t ws_size,
                              hipStream_t stream)
{
  (void)in_sizes; (void)n_in; (void)out_size;
  const float* enc       = (const float*)d_in[0];
  const int*   caps      = (const int*)  d_in[1];
  const int*   clen      = (const int*)  d_in[2];
  const float* emb       = (const float*)d_in[3];
  const float* W_enc_att = (const float*)d_in[4];
  const float* b_enc_att = (const float*)d_in[5];
  const float* W_dec_att = (const float*)d_in[6];
  const float* b_dec_att = (const float*)d_in[7];
  const float* W_full_att= (const float*)d_in[8];
  const float* b_full_att= (const float*)d_in[9];
  const float* W_init_h  = (const float*)d_in[10];
  const float* b_init_h  = (const float*)d_in[11];
  const float* W_init_c  = (const float*)d_in[12];
  const float* b_init_c  = (const float*)d_in[13];
  const float* W_f_beta  = (const float*)d_in[14];
  const float* b_f_beta  = (const float*)d_in[15];
  const float* W_fc      = (const float*)d_in[16];
  const float* b_fc      = (const float*)d_in[17];
  const float* W_ih      = (const float*)d_in[18];
  const float* W_hh      = (const float*)d_in[19];
  const float* b_ih      = (const float*)d_in[20];
  const float* b_hh      = (const float*)d_in[21];
  float* out = (float*)d_out;

  char* ws = (char*)d_ws;
  size_t off = 0;
  auto take = [&](size_t bytes)->char*{
    char* p = ws + off; off = (off + bytes + 255) & ~(size_t)255; return p; };
  int*   sort_ind = (int*)  take((size_t)B_*4);
  int*   dec_len  = (int*)  take((size_t)B_*4);
  int*   rowmap   = (int*)  take((size_t)B_*P_*4);
  float* mean_enc = (float*)take((size_t)B_*ENC_*4);
  float* h        = (float*)take((size_t)B_*DEC_*4);
  float* c        = (float*)take((size_t)B_*DEC_*4);
  float* att2     = (float*)take((size_t)B_*ATT_*4);
  float* scores   = (float*)take((size_t)B_*P_*4);
  float* alpha    = (float*)take((size_t)B_*P_*4);
  float* awe      = (float*)take((size_t)B_*ENC_*4);
  float* gpre     = (float*)take((size_t)B_*ENC_*4);
  float* x        = (float*)take((size_t)B_*XD_*4);
  float* gates    = (float*)take((size_t)B_*4*DEC_*4);
  float* att1     = (float*)take((size_t)B_*P_*ATT_*4);
  if (off > ws_size) return;   // workspace too small; refuse deterministically

  dim3 blk(256);
  sort_kernel  <<<1, B_, 0, stream>>>(clen, sort_ind, dec_len);
  rowmap_kernel<<<(B_*P_ + 255)/256, blk, 0, stream>>>(sort_ind, rowmap);
  mean_kernel  <<<dim3(ENC_/256, B_), blk, 0, stream>>>(enc, sort_ind, mean_enc);

  // h0 / c0: (256 x 2048) * (512 x 2048)^T
  gemm_wmma_f32<0><<<dim3(DEC_/64, B_/64), blk, 0, stream>>>(
      mean_enc, W_init_h, b_init_h, h, B_, DEC_, ENC_, DEC_, 0, nullptr, nullptr, 0);
  gemm_wmma_f32<0><<<dim3(DEC_/64, B_/64), blk, 0, stream>>>(
      mean_enc, W_init_c, b_init_c, c, B_, DEC_, ENC_, DEC_, 0, nullptr, nullptr, 0);

  // att1: (50176 x 2048) * (512 x 2048)^T, A rows gathered through sort
  gemm_wmma_f32<0><<<dim3(ATT_/64, (B_*P_)/64), blk, 0, stream>>>(
      enc, W_enc_att, b_enc_att, att1, B_*P_, ATT_, ENC_, ATT_, 0, rowmap, nullptr, 0);

  for (int t = 0; t < T_; ++t){
    gemm_wmma_f32<0><<<dim3(ATT_/64, B_/64), blk, 0, stream>>>(
        h, W_dec_att, b_dec_att, att2, B_, ATT_, DEC_, ATT_, 0, nullptr, nullptr, 0);
    scores_kernel <<<(B_*P_)/8, blk, 0, stream>>>(att1, att2, W_full_att, b_full_att, scores);
    softmax_kernel<<<B_, blk, 0, stream>>>(scores, alpha);
    awe_kernel    <<<dim3(ENC_/256, B_), blk, 0, stream>>>(enc, alpha, sort_ind, awe);
    gemm_wmma_f32<0><<<dim3(ENC_/64, B_/64), blk, 0, stream>>>(
        h, W_f_beta, b_f_beta, gpre, B_, ENC_, DEC_, ENC_, 0, nullptr, nullptr, 0);
    build_x_kernel<<<dim3(XD_/256, B_), blk, 0, stream>>>(caps, sort_ind, emb, gpre, awe, x, t);
    gemm_wmma_f32<0><<<dim3((4*DEC_)/64, B_/64), blk, 0, stream>>>(
        x, W_ih, b_ih, gates, B_, 4*DEC_, XD_, 4*DEC_, 0, nullptr, nullptr, 0);
    gemm_wmma_f32<1><<<dim3((4*DEC_)/64, B_/64), blk, 0, stream>>>(
        h, W_hh, b_hh, gates, B_, 4*DEC_, DEC_, 4*DEC_, 0, nullptr, nullptr, 0);
    lstm_kernel<<<dim3(DEC_/256, B_), blk, 0, stream>>>(gates, h, c, dec_len, t);
    // vocab projection, masked, strided store into (B, T, V) output
    gemm_wmma_f32<2><<<dim3((V_ + 63)/64, B_/64), blk, 0, stream>>>(
        h, W_fc, b_fc, out, B_, V_, DEC_, (long)T_*V_, (long)t*V_, nullptr, dec_len, t);
  }
}